// VectorQuantizedEmbeddings_49435073577018
// MI455X (gfx1250) — compile-verified
//
#include <hip/hip_runtime.h>
#include <hip/hip_bf16.h>

typedef __attribute__((ext_vector_type(2))) float v2f;
typedef __attribute__((ext_vector_type(8))) float v8f;

#define NROWS   16384   // 8 * 2048 tokens
#define NCODES  8192
#define DIM     256
#define CSPLIT  4                 // code-range split across blocks
#define CODES_PER_CHUNK (NCODES / CSPLIT)   // 2048
#define TILES_PER_CHUNK (CODES_PER_CHUNK / 16) // 128
#define ROWS_PER_BLOCK  128       // 8 waves * 16 rows
#define LDSROW  260               // 256 + 4 pad -> conflict-free strided reads
#define QELEMS  ((size_t)NROWS * DIM)

// ---------------- kernel 1: e_sq[k] = ||embedding_k||^2 -----------------
__global__ void vq_esq_kernel(const float* __restrict__ emb,
                              float* __restrict__ esq) {
    const int lane = threadIdx.x & 31;
    const int code = blockIdx.x * 8 + (threadIdx.x >> 5);
    const float4* e4 = (const float4*)(emb + (size_t)code * DIM);
    float4 a = e4[lane];
    float4 b = e4[lane + 32];
    float s = a.x * a.x + a.y * a.y + a.z * a.z + a.w * a.w
            + b.x * b.x + b.y * b.y + b.z * b.z + b.w * b.w;
    #pragma unroll
    for (int m = 16; m >= 1; m >>= 1) s += __shfl_xor(s, m, 32);
    if (lane == 0) esq[code] = s;
}

// -------- kernel 2: WMMA f32 distance scan with running argmin ----------
__global__ void __launch_bounds__(256)
vq_argmin_kernel(const float* __restrict__ z,
                 const float* __restrict__ emb,
                 const float* __restrict__ eSq,
                 float* __restrict__ partVal,
                 int*   __restrict__ partIdx) {
    __shared__ float eTile[16 * LDSROW];   // 16 codes x 256 dims (padded)

    const int t     = threadIdx.x;
    const int lane  = t & 31;
    const int wave  = t >> 5;
    const int chunk = blockIdx.x % CSPLIT;
    const int rowBase = (blockIdx.x / CSPLIT) * ROWS_PER_BLOCK + wave * 16;

    // ---- A fragments: 16 rows x 256 dims resident in 128 VGPRs ----
    // A layout (32-bit, 16x4): lanes 0-15 hold K=0,1 ; lanes 16-31 hold K=2,3
    const float* zrow = z + (size_t)(rowBase + (lane & 15)) * DIM + ((lane >> 4) << 1);
    v2f af[64];
    #pragma unroll
    for (int kk = 0; kk < 64; ++kk)
        af[kk] = *(const v2f*)&zrow[kk << 2];

    // B fragment base in LDS: col = lane&15 (code), K offset 2*(lane>>4)
    const int bBase = (lane & 15) * LDSROW + ((lane >> 4) << 1);

    float minv[8];
    int   mini[8];
    #pragma unroll
    for (int r = 0; r < 8; ++r) { minv[r] = 3.4e38f; mini[r] = 0; }

    const v8f vzero = {};

    for (int ct = 0; ct < TILES_PER_CHUNK; ++ct) {
        const int c0 = chunk * CODES_PER_CHUNK + ct * 16;

        __syncthreads();
        // cooperative load of 16 embedding rows (16*64 float4 chunks)
        #pragma unroll
        for (int j = t; j < 1024; j += 256) {
            const int code = j >> 6;
            const int c4   = j & 63;
            const float4 v = *(const float4*)(emb + (size_t)(c0 + code) * DIM + (c4 << 2));
            *(float4*)&eTile[code * LDSROW + (c4 << 2)] = v;
        }
        __syncthreads();

        // ---- 64 x V_WMMA_F32_16X16X4_F32, 4 accumulators to hide latency
        v8f a0 = vzero, a1 = vzero, a2 = vzero, a3 = vzero;
        #pragma unroll
        for (int kk = 0; kk < 64; kk += 4) {
            v2f b0 = *(const v2f*)&eTile[bBase + ((kk + 0) << 2)];
            a0 = __builtin_amdgcn_wmma_f32_16x16x4_f32(false, af[kk + 0], false, b0, (short)0, a0, false, false);
            v2f b1 = *(const v2f*)&eTile[bBase + ((kk + 1) << 2)];
            a1 = __builtin_amdgcn_wmma_f32_16x16x4_f32(false, af[kk + 1], false, b1, (short)0, a1, false, false);
            v2f b2 = *(const v2f*)&eTile[bBase + ((kk + 2) << 2)];
            a2 = __builtin_amdgcn_wmma_f32_16x16x4_f32(false, af[kk + 2], false, b2, (short)0, a2, false, false);
            v2f b3 = *(const v2f*)&eTile[bBase + ((kk + 3) << 2)];
            a3 = __builtin_amdgcn_wmma_f32_16x16x4_f32(false, af[kk + 3], false, b3, (short)0, a3, false, false);
        }

        // ---- running argmin update: score = e_sq - 2*dot (||z||^2 invariant)
        const float esq   = eSq[c0 + (lane & 15)];
        const int codeId  = c0 + (lane & 15);
        #pragma unroll
        for (int r = 0; r < 8; ++r) {
            const float dot = (a0[r] + a1[r]) + (a2[r] + a3[r]);
            const float score = fmaf(-2.0f, dot, esq);
            const bool take = score < minv[r];
            minv[r] = take ? score : minv[r];
            mini[r] = take ? codeId : mini[r];
        }
    }

    // ---- cross-lane tournament within each 16-lane half (rows r / r+8) ----
    #pragma unroll
    for (int r = 0; r < 8; ++r) {
        float v = minv[r];
        int   i = mini[r];
        #pragma unroll
        for (int m = 8; m >= 1; m >>= 1) {
            const float ov = __shfl_xor(v, m, 32);
            const int   oi = __shfl_xor(i, m, 32);
            const bool take = (ov < v) || (ov == v && oi < i);
            v = take ? ov : v;
            i = take ? oi : i;
        }
        if ((lane & 15) == 0) {
            const int row = rowBase + r + ((lane >> 4) << 3);
            partVal[(size_t)row * CSPLIT + chunk] = v;
            partIdx[(size_t)row * CSPLIT + chunk] = i;
        }
    }
}

// -------- kernel 3: combine code-chunk partials, emit token ids ---------
__global__ void vq_combine_kernel(const float* __restrict__ partVal,
                                  const int*   __restrict__ partIdx,
                                  int*   __restrict__ ids,
                                  float* __restrict__ out_ids_f32) {
    const int row = blockIdx.x * blockDim.x + threadIdx.x;
    if (row >= NROWS) return;
    float bv = 3.4e38f;
    int   bi = 0;
    #pragma unroll
    for (int c = 0; c < CSPLIT; ++c) {     // ascending chunk = ascending code idx
        const float v = partVal[(size_t)row * CSPLIT + c];
        const int   i = partIdx[(size_t)row * CSPLIT + c];
        if (v < bv) { bv = v; bi = i; }    // strict < keeps first occurrence
    }
    ids[row] = bi;
    out_ids_f32[row] = (float)bi;
}

// -------- kernel 4: gather quantized = embedding[token_id] --------------
__global__ void vq_gather_kernel(const float* __restrict__ emb,
                                 const int* __restrict__ ids,
                                 float* __restrict__ outq) {
    const int row = blockIdx.x;
    const int id  = ids[row];
    const float4* src = (const float4*)(emb + (size_t)id * DIM);
    float4* dst = (float4*)(outq + (size_t)row * DIM);
    dst[threadIdx.x] = src[threadIdx.x];
}

extern "C" void kernel_launch(void* const* d_in, const int* in_sizes, int n_in,
                              void* d_out, int out_size, void* d_ws, size_t ws_size,
                              hipStream_t stream) {
    const float* z   = (const float*)d_in[0];   // (8,2048,256) f32
    const float* emb = (const float*)d_in[1];   // (8192,256) f32
    float* out = (float*)d_out;                 // quantized (NROWS*DIM) then ids (NROWS)

    // workspace layout
    char* ws = (char*)d_ws;
    float* eSq     = (float*)ws;                                   // 8192 f32
    float* partVal = (float*)(ws + 32 * 1024);                     // NROWS*CSPLIT f32
    int*   partIdx = (int*)  (ws + 32 * 1024 + NROWS * CSPLIT * 4);// NROWS*CSPLIT i32
    int*   ids     = (int*)  (ws + 32 * 1024 + NROWS * CSPLIT * 8);// NROWS i32

    vq_esq_kernel<<<NCODES / 8, 256, 0, stream>>>(emb, eSq);

    vq_argmin_kernel<<<(NROWS / ROWS_PER_BLOCK) * CSPLIT, 256, 0, stream>>>(
        z, emb, eSq, partVal, partIdx);

    vq_combine_kernel<<<(NROWS + 255) / 256, 256, 0, stream>>>(
        partVal, partIdx, ids, out + QELEMS);

    vq_gather_kernel<<<NROWS, DIM / 4, 0, stream>>>(emb, ids, out);
}